// NewtonSOR_22479858827500
// MI455X (gfx1250) — compile-verified
//
#include <hip/hip_runtime.h>

#define NVARS 128
#define STR   129      // padded LDS row stride (floats) -> conflict-free column reads
#define BLOCK 256      // 8 waves: 8 row-blocks of 16 for the WMMA matvec
#define MAXIT 16

typedef float v2f __attribute__((ext_vector_type(2)));
typedef float v8f __attribute__((ext_vector_type(8)));
typedef unsigned int v4u __attribute__((ext_vector_type(4)));
typedef unsigned int v8u __attribute__((ext_vector_type(8)));

__global__ __launch_bounds__(BLOCK) void newton_sor_kernel(
    const float* __restrict__ x_in,
    const float* __restrict__ A,
    const float* __restrict__ b_in,
    const float* __restrict__ omega_in,
    float* __restrict__ x_out)
{
    __shared__ float As[NVARS * STR];   // 66048 B, row stride 129
    __shared__ float xs[NVARS];
    __shared__ float Fs[NVARS];
    __shared__ float dxs[NVARS];
    __shared__ float invd[NVARS];
    __shared__ float ys[NVARS];

    const int tid  = threadIdx.x;
    const int lane = tid & 31;
    const int wid  = tid >> 5;
    const int bid  = blockIdx.x;

    const float omega = omega_in[bid];
    const float* Ab = A + (size_t)bid * NVARS * NVARS;

    // ---- Tensor Data Mover: DMA the 128x128 f32 A tile into LDS, padding
    // each 128-dword row with 1 dword (LDS stride = 129 floats). One wave issues.
    if (wid == 0) {
        unsigned int lds_base = (unsigned int)(size_t)(&As[0]); // LDS offset = addr[31:0]
        unsigned long long ga = (unsigned long long)(size_t)Ab;
        v4u g0;
        g0[0] = 1u;                                  // count=1 (valid), user mode
        g0[1] = lds_base;                            // lds_addr
        g0[2] = (unsigned int)ga;                    // global_addr[31:0]
        g0[3] = ((unsigned int)(ga >> 32) & 0x1FFFFFFu) | (2u << 30); // addr[56:32] | type=2
        v8u g1;
        g1[0] = (2u << 16)        // data_size = 4 bytes
              | (1u << 20)        // pad_enable
              | (6u << 22)        // pad_interval: every 128 dwords
              | (0u << 25);       // pad_amount: 1 dword
        g1[1] = ((unsigned)NVARS) << 16;  // tensor_dim0[15:0] @ bits 63:48
        g1[2] = ((unsigned)NVARS) << 16;  // tensor_dim1[15:0] @ bits 111:96.. (bit80 base)
        g1[3] = ((unsigned)NVARS) << 16;  // tile_dim0 @ bits 127:112
        g1[4] = (unsigned)NVARS;          // tile_dim1 @ bits 143:128, tile_dim2=0
        g1[5] = (unsigned)NVARS;          // tensor_dim0_stride[31:0]
        g1[6] = 0u;
        g1[7] = 0u;
        v4u gz = (v4u){0u, 0u, 0u, 0u};   // groups 2/3: dims/tiles beyond 2D unused
        asm volatile("tensor_load_to_lds %0, %1, %2, %3"
                     :: "s"(g0), "s"(g1), "s"(gz), "s"(gz)
                     : "memory");
        __builtin_amdgcn_s_wait_tensorcnt(0);
    }

    // per-row state in registers of threads 0..127
    float xi = 0.0f, bi = 0.0f;
    if (tid < NVARS) {
        xi = x_in[(size_t)bid * NVARS + tid];
        bi = b_in[(size_t)bid * NVARS + tid];
        xs[tid] = xi;
    }
    __syncthreads();

    // ---- f32 WMMA matvec: ys = As(128x128) * xs(128). 8 waves, one 16-row block each.
    auto matvec = [&]() {
        const int rb  = wid;                       // row block 0..7
        const int row = (rb << 4) + (lane & 15);
        const int colSel = (lane >> 4) << 1;       // lanes 0-15: K=0,1; lanes 16-31: K=2,3
        v8f c = {};
        for (int k0 = 0; k0 < NVARS; k0 += 4) {
            v2f a;
            a[0] = As[row * STR + k0 + colSel];
            a[1] = As[row * STR + k0 + colSel + 1];
            float x0 = xs[k0 + 0], x1 = xs[k0 + 1];
            float x2 = xs[k0 + 2], x3 = xs[k0 + 3];
            v2f bf;                                 // B: x in column N=0, rest zero
            bf[0] = (lane == 0) ? x0 : ((lane == 16) ? x2 : 0.0f);
            bf[1] = (lane == 0) ? x1 : ((lane == 16) ? x3 : 0.0f);
            c = __builtin_amdgcn_wmma_f32_16x16x4_f32(
                    false, a, false, bf, (short)0, c, false, false);
        }
        if ((lane & 15) == 0) {                    // column N=0 lives in lanes 0 and 16
            int base = (rb << 4) + ((lane >> 4) << 3);
            #pragma unroll
            for (int m = 0; m < 8; ++m) ys[base + m] = c[m];
        }
    };

    // F = A x + x^3 - b
    matvec();
    __syncthreads();
    if (tid < NVARS) Fs[tid] = ys[tid] + xi * xi * xi - bi;
    __syncthreads();

    for (int it = 0; it < MAXIT; ++it) {
        if (tid < NVARS) invd[tid] = 1.0f / (As[tid * STR + tid] + 3.0f * xi * xi);
        __syncthreads();

        // ---- forward substitution (M = D + omega*strict_lower(A)) in wave 0,
        // barrier-free: lane l owns rows l, l+32, l+64, l+96.
        if (wid == 0) {
            float acc[4];
            #pragma unroll
            for (int k = 0; k < 4; ++k) acc[k] = Fs[(k << 5) + lane];
            #pragma unroll
            for (int k = 0; k < 4; ++k) {
                for (int jj = 0; jj < 32; ++jj) {
                    int j = (k << 5) + jj;
                    float dxj = __shfl(acc[k], jj, 32) * invd[j];
                    if (lane == jj) dxs[j] = dxj;
                    float w = omega * dxj;
                    if (lane > jj) acc[k] -= w * As[((k << 5) + lane) * STR + j];
                    #pragma unroll
                    for (int k2 = k + 1; k2 < 4; ++k2)
                        acc[k2] -= w * As[((k2 << 5) + lane) * STR + j];
                }
            }
        }
        __syncthreads();

        if (tid < NVARS) { xi -= omega * dxs[tid]; xs[tid] = xi; }
        __syncthreads();

        matvec();
        __syncthreads();
        if (tid < NVARS) Fs[tid] = ys[tid] + xi * xi * xi - bi;
        __syncthreads();
    }

    if (tid < NVARS) x_out[(size_t)bid * NVARS + tid] = xi;
}

extern "C" void kernel_launch(void* const* d_in, const int* in_sizes, int n_in,
                              void* d_out, int out_size, void* d_ws, size_t ws_size,
                              hipStream_t stream) {
    const float* x     = (const float*)d_in[0];
    const float* A     = (const float*)d_in[1];
    const float* b     = (const float*)d_in[2];
    const float* omega = (const float*)d_in[3];
    float* out = (float*)d_out;
    int batch = in_sizes[0] / NVARS;   // 2048
    newton_sor_kernel<<<batch, BLOCK, 0, stream>>>(x, A, b, omega, out);
}